// TASS_90787018703635
// MI455X (gfx1250) — compile-verified
//
#include <hip/hip_runtime.h>
#include <math.h>

typedef float v2f __attribute__((ext_vector_type(2)));
typedef float v8f __attribute__((ext_vector_type(8)));

#define CC   2048
#define LL   4096
#define L2   2048
#define W3   6144   /* conv1_w row stride = C*3 */

/* workspace layout (float offsets) */
#define WS_CONST   0                 /* [4]=meanI [5]=denomI [6]=meanS [7]=denomS [8]=Bsum */
#define WS_RSUMI   16
#define WS_RSQI    (WS_RSUMI + 2048)
#define WS_RSUMS   (WS_RSQI  + 2048)
#define WS_RSQS    (WS_RSUMS + 2048)
#define WS_COLSUM  (WS_RSQS  + 2048)
#define WS_T       (WS_COLSUM+ 2048)
#define WS_U       (WS_T     + 2048)
#define WS_V       (WS_U     + 6144)
#define WS_SPAD    (WS_V     + 6144) /* 2050 used (1 zero halo each side), reserve 2064 */
#define WS_OUT2    (WS_SPAD  + 2064)
#define WS_Q       (WS_OUT2  + 2048)
#define WS_OUT1    (WS_Q     + 6144)
#define WS_MIX     (WS_OUT1  + 2048)
#define WS_FIN     (WS_MIX   + 2048)
#define WS_TOTAL   (WS_FIN   + 2048)

__device__ __forceinline__ float sigm(float x) { return 1.0f / (1.0f + expf(-x)); }

__device__ __forceinline__ float block_reduce256(float v, float* sh) {
    int t = threadIdx.x;
    sh[t] = v; __syncthreads();
    for (int s = 128; s > 0; s >>= 1) { if (t < s) sh[t] += sh[t + s]; __syncthreads(); }
    float r = sh[0]; __syncthreads();
    return r;
}

/* pass 1: per-row sum & sumsq for input (blocks 0..2047) and insum (2048..4095) */
__global__ void k_row_stats(const float* __restrict__ input, const float* __restrict__ insum,
                            float* __restrict__ ws) {
    int b = blockIdx.x;
    const float* src = (b < CC) ? input + (size_t)b * LL : insum + (size_t)(b - CC) * LL;
    const float4* s4 = (const float4*)src;
    float sum = 0.f, ssq = 0.f;
    for (int j = threadIdx.x; j < LL / 4; j += 256) {
        float4 f = s4[j];
        sum += f.x + f.y + f.z + f.w;
        ssq += f.x * f.x + f.y * f.y + f.z * f.z + f.w * f.w;
    }
    __shared__ float sh[256];
    sum = block_reduce256(sum, sh);
    ssq = block_reduce256(ssq, sh);
    if (threadIdx.x == 0) {
        if (b < CC) { ws[WS_RSUMI + b] = sum; ws[WS_RSQI + b] = ssq; }
        else        { ws[WS_RSUMS + (b - CC)] = sum; ws[WS_RSQS + (b - CC)] = ssq; }
    }
}

/* reduce row stats -> simam constants; also Bsum = sum(fc_b) */
__global__ void k_reduce_stats(const float* __restrict__ fcb, float* __restrict__ ws) {
    float a0 = 0, a1 = 0, a2 = 0, a3 = 0, a4 = 0;
    for (int i = threadIdx.x; i < CC; i += 256) {
        a0 += ws[WS_RSUMI + i]; a1 += ws[WS_RSQI + i];
        a2 += ws[WS_RSUMS + i]; a3 += ws[WS_RSQS + i];
        a4 += fcb[i];
    }
    __shared__ float sh[256];
    a0 = block_reduce256(a0, sh); a1 = block_reduce256(a1, sh);
    a2 = block_reduce256(a2, sh); a3 = block_reduce256(a3, sh);
    a4 = block_reduce256(a4, sh);
    if (threadIdx.x == 0) {
        const float N = (float)CC * (float)LL;
        float meanI = a0 / N;
        float dI    = a1 - a0 * a0 / N;
        float denI  = 4.0f * (dI / (N - 1.0f) + 1e-4f);
        float meanS = a2 / N;
        float dS    = a3 - a2 * a2 / N;
        float denS  = 4.0f * (dS / (N - 1.0f) + 1e-4f);
        ws[4] = meanI; ws[5] = denI; ws[6] = meanS; ws[7] = denS; ws[8] = a4;
    }
}

/* colsum_fc[i] = sum_c fc_w[c,i] */
__global__ void k_colsum(const float* __restrict__ fcw, float* __restrict__ ws) {
    int i  = blockIdx.x * 256 + threadIdx.x;
    int c0 = blockIdx.y * 64;
    float s = 0.f;
    for (int c = c0; c < c0 + 64; ++c) s += fcw[(size_t)c * CC + i];
    atomicAdd(&ws[WS_COLSUM + i], s);
}

/* t[c] = fc_w[c,:] . (0.5*rowsum_input) + 2048*fc_b[c] */
__global__ void k_t(const float* __restrict__ fcw, const float* __restrict__ fcb,
                    float* __restrict__ ws) {
    int c = blockIdx.x;
    float acc = 0.f;
    const float* row = fcw + (size_t)c * CC;
    for (int i = threadIdx.x; i < CC; i += 256) acc += row[i] * (0.5f * ws[WS_RSUMI + i]);
    __shared__ float sh[256];
    acc = block_reduce256(acc, sh);
    if (threadIdx.x == 0) ws[WS_T + c] = acc + fcb[c] * 2048.0f;
}

/* u[m] = sum_i colsum[i]*w[i,m]; v[m] = sum_i t[i]*w[i,m]   (m = j*3+k) */
__global__ void k_uv(const float* __restrict__ w, float* __restrict__ ws) {
    int m  = blockIdx.x * 256 + threadIdx.x;
    int i0 = blockIdx.y * 64;
    float su = 0.f, sv = 0.f;
    for (int i = i0; i < i0 + 64; ++i) {
        float wv = w[(size_t)i * W3 + m];
        su += ws[WS_COLSUM + i] * wv;
        sv += ws[WS_T + i] * wv;
    }
    atomicAdd(&ws[WS_U + m], su);
    atomicAdd(&ws[WS_V + m], sv);
}

/* s_raw[l] = sum_{j,k} u[j,k]*xm[j,l+k-1]; out2pre[l] same with v; xm = avgpool2(insum) */
__global__ void k_s_out2(const float* __restrict__ insum, float* __restrict__ ws) {
    __shared__ float sh[258];
    int l0 = blockIdx.x * 256;
    int j0 = blockIdx.y * 64;
    int tid = threadIdx.x;
    int l = l0 + tid;
    float sacc = 0.f, oacc = 0.f;
    for (int j = j0; j < j0 + 64; ++j) {
        const float2* row2 = (const float2*)(insum + (size_t)j * LL);
        float2 p = row2[l];
        sh[tid + 1] = 0.5f * (p.x + p.y);
        if (tid == 0) {
            float left = 0.f;
            if (l0 > 0) { float2 q = row2[l0 - 1]; left = 0.5f * (q.x + q.y); }
            sh[0] = left;
        }
        if (tid == 1) {
            float right = 0.f;
            int rc = l0 + 256;
            if (rc < L2) { float2 q = row2[rc]; right = 0.5f * (q.x + q.y); }
            sh[257] = right;
        }
        __syncthreads();
        float u0 = ws[WS_U + 3 * j], u1 = ws[WS_U + 3 * j + 1], u2 = ws[WS_U + 3 * j + 2];
        float v0 = ws[WS_V + 3 * j], v1 = ws[WS_V + 3 * j + 1], v2 = ws[WS_V + 3 * j + 2];
        float xl = sh[tid], xc = sh[tid + 1], xr = sh[tid + 2];
        sacc += u0 * xl + u1 * xc + u2 * xr;
        oacc += v0 * xl + v1 * xc + v2 * xr;
        __syncthreads();
    }
    atomicAdd(&ws[WS_SPAD + 1 + l], sacc);
    atomicAdd(&ws[WS_OUT2 + l], oacc);
}

/* add bias B to interior of sPad (halo stays zero) */
__global__ void k_fin_s(float* __restrict__ ws) {
    int l = blockIdx.x * 256 + threadIdx.x;
    if (l < L2) ws[WS_SPAD + 1 + l] += ws[8];
}

/* q[i,k] = sum_l x[i,l]*sPad[l-k+2] via WMMA f32 16x16x4.
   A tile 16x4: lane holds rows m=lane&15, K pair at 2*(lane>>4).
   B tile 4x16: b.x=B[2*hi,n], b.y=B[2*hi+1,n], n=lane&15; B[kk,n]=sPad[kk_abs-n+2] for n<3.
   D: vgpr r, lane -> D[r+8*hi, n]. One wave (32 thr) per block; EXEC all ones at WMMA. */
__global__ void k_q_wmma(const float* __restrict__ input, float* __restrict__ ws) {
    const float* sPad = ws + WS_SPAD;
    int lane = threadIdx.x;
    int n  = lane & 15;
    int hi = lane >> 4;
    int i0  = blockIdx.x * 16;
    int lg0 = blockIdx.y * 256;
    int row = i0 + n;                 /* A-fragment row m == lane&15 */
    float msk = (n < 3) ? 1.0f : 0.0f;
    v8f acc = {0.f, 0.f, 0.f, 0.f, 0.f, 0.f, 0.f, 0.f};
    for (int kk = 0; kk < 256; kk += 4) {
        int lg = lg0 + kk;
        /* pooled A values: x[row, lg+2*hi], x[row, lg+2*hi+1] from 4 input floats */
        const float4 f = *(const float4*)(input + (size_t)row * LL + 2 * lg + 4 * hi);
        v2f a; a.x = 0.5f * (f.x + f.y); a.y = 0.5f * (f.z + f.w);
        int kA = lg + 2 * hi;
        int ix = kA - n + 2;     ix = ix < 0 ? 0 : ix;
        int iy = kA + 1 - n + 2; iy = iy < 0 ? 0 : iy;
        v2f b; b.x = sPad[ix] * msk; b.y = sPad[iy] * msk;
        acc = __builtin_amdgcn_wmma_f32_16x16x4_f32(false, a, false, b,
                                                    (short)0, acc, false, false);
    }
    if (n < 3) {
        for (int r = 0; r < 8; ++r)
            atomicAdd(&ws[WS_Q + (size_t)(i0 + r + 8 * hi) * 3 + n], acc[r]);
    }
}

/* out1pre[o] = w[o,:,:] (flat 6144) . q */
__global__ void k_out1(const float* __restrict__ w, float* __restrict__ ws) {
    int o = blockIdx.x;
    const float* wo = w + (size_t)o * W3;
    float acc = 0.f;
    for (int idx = threadIdx.x; idx < W3; idx += 256) acc += wo[idx] * ws[WS_Q + idx];
    __shared__ float sh[256];
    acc = block_reduce256(acc, sh);
    if (threadIdx.x == 0) ws[WS_OUT1 + o] = acc;
}

__global__ void k_mix(const float* __restrict__ mixw, float* __restrict__ ws) {
    int i = blockIdx.x * 256 + threadIdx.x;
    if (i < CC) {
        float mf = sigm(mixw[0]);
        ws[WS_MIX + i] = sigm(ws[WS_OUT1 + i]) * mf + sigm(ws[WS_OUT2 + i]) * (1.0f - mf);
    }
}

/* final[o] = sigmoid( sum_i w[o,i,1] * mix[i] )   (length-1 conv, pad=1 -> middle tap) */
__global__ void k_final_mv(const float* __restrict__ w, float* __restrict__ ws) {
    int o = blockIdx.x;
    const float* wo = w + (size_t)o * W3;
    float acc = 0.f;
    for (int i = threadIdx.x; i < CC; i += 256) acc += wo[i * 3 + 1] * ws[WS_MIX + i];
    __shared__ float sh[256];
    acc = block_reduce256(acc, sh);
    if (threadIdx.x == 0) ws[WS_FIN + o] = sigm(acc);
}

__device__ __forceinline__ float simam1(float x, float mean, float den) {
    float d = x - mean;
    float y = d * d / den + 0.5f;
    return x * (1.0f + sigm(y));
}

/* out[c,l] = (simam(input)*0.6 + simam(insum)*0.4) * final[c] */
__global__ void k_elem(const float* __restrict__ input, const float* __restrict__ insum,
                       const float* __restrict__ ws, float* __restrict__ out) {
    size_t g = (size_t)blockIdx.x * 256 + threadIdx.x;     /* float4 index */
    float meanI = ws[4], denI = ws[5], meanS = ws[6], denS = ws[7];
    int c = (int)(g >> 10);                                /* (g*4)/4096 */
    float fin = ws[WS_FIN + c];
    float4 a = ((const float4*)input)[g];
    float4 b = ((const float4*)insum)[g];
    float4 r;
    r.x = (simam1(a.x, meanI, denI) * 0.6f + simam1(b.x, meanS, denS) * 0.4f) * fin;
    r.y = (simam1(a.y, meanI, denI) * 0.6f + simam1(b.y, meanS, denS) * 0.4f) * fin;
    r.z = (simam1(a.z, meanI, denI) * 0.6f + simam1(b.z, meanS, denS) * 0.4f) * fin;
    r.w = (simam1(a.w, meanI, denI) * 0.6f + simam1(b.w, meanS, denS) * 0.4f) * fin;
    ((float4*)out)[g] = r;
}

extern "C" void kernel_launch(void* const* d_in, const int* in_sizes, int n_in,
                              void* d_out, int out_size, void* d_ws, size_t ws_size,
                              hipStream_t stream) {
    const float* input = (const float*)d_in[0];
    const float* insum = (const float*)d_in[1];
    const float* w     = (const float*)d_in[2];   /* conv1_w (C,C,3) */
    const float* fcw   = (const float*)d_in[3];   /* fc_w (C,C,1) */
    const float* fcb   = (const float*)d_in[4];   /* fc_b (C,) */
    const float* mixw  = (const float*)d_in[5];   /* mix_w (1,) */
    float* out = (float*)d_out;
    float* ws  = (float*)d_ws;
    (void)in_sizes; (void)n_in; (void)out_size; (void)ws_size;

    hipMemsetAsync(ws, 0, (size_t)WS_TOTAL * sizeof(float), stream);

    k_row_stats   <<<dim3(2 * CC),    256, 0, stream>>>(input, insum, ws);
    k_reduce_stats<<<dim3(1),         256, 0, stream>>>(fcb, ws);
    k_colsum      <<<dim3(8, 32),     256, 0, stream>>>(fcw, ws);
    k_t           <<<dim3(CC),        256, 0, stream>>>(fcw, fcb, ws);
    k_uv          <<<dim3(24, 32),    256, 0, stream>>>(w, ws);
    k_s_out2      <<<dim3(8, 32),     256, 0, stream>>>(insum, ws);
    k_fin_s       <<<dim3(8),         256, 0, stream>>>(ws);
    k_q_wmma      <<<dim3(128, 8),     32, 0, stream>>>(input, ws);
    k_out1        <<<dim3(CC),        256, 0, stream>>>(w, ws);
    k_mix         <<<dim3(8),         256, 0, stream>>>(mixw, ws);
    k_final_mv    <<<dim3(CC),        256, 0, stream>>>(w, ws);
    k_elem        <<<dim3(8192),      256, 0, stream>>>(input, insum, ws, out);
}